// S2DPnP_69784628625926
// MI455X (gfx1250) — compile-verified
//
#include <hip/hip_runtime.h>
#include <math.h>

#define CC 128
#define HH 512
#define WW 512
#define EPSF 1e-12f

typedef __attribute__((ext_vector_type(2))) float v2f;
typedef __attribute__((ext_vector_type(8))) float v8f;

// ---- workspace layout (floats) ----
#define WS_R    0    // 9: current R
#define WS_T    9    // 3: current t
#define WS_LAM  12   // 1
#define WS_COST 13   // 1: cost_best
#define WS_JAC  14   // 6: jacobi
#define WS_H    20   // 49: 7x7 normal-equation accumulator (unscaled)
#define WS_CNT  69   // 1: sum vf
#define WS_NUM  70   // 1: trial cost numerator
#define WS_DEN  71   // 1: trial cost denominator
#define WS_RC   72   // 9: candidate R
#define WS_TC   81   // 3: candidate t
#define WS_FEAT 128  // transposed features (H,W,C)

__device__ __forceinline__ float wave_sum(float v) {
#pragma unroll
  for (int s = 16; s > 0; s >>= 1) v += __shfl_xor(v, s, 32);
  return v;
}

// cs = channel stride, ps = pixel stride (CHW: cs=H*W, ps=1; HWC: cs=1, ps=C)
__device__ __forceinline__ float fetchf(const float* f, long cs, long ps,
                                        int c, int y, int x) {
  return f[(long)c * cs + ((long)y * WW + x) * ps];
}

__device__ __forceinline__ void proj_pt(const float* R, const float* t,
                                        float fx, float fy, float cx, float cy,
                                        float X, float Y, float Z,
                                        float& px, float& py, float& qx, float& qy,
                                        float& qz, int& inim) {
  qx = R[0]*X + R[1]*Y + R[2]*Z + t[0];
  qy = R[3]*X + R[4]*Y + R[5]*Z + t[1];
  qz = R[6]*X + R[7]*Y + R[8]*Z + t[2];
  float invd = 1.f / qz;
  px = qx * invd * fx + cx;
  py = qy * invd * fy + cy;
  inim = (px >= 1.f && px < (float)(WW - 1) && py >= 1.f && py < (float)(HH - 1)) ? 1 : 0;
}

// whole-wave: ||f-fref||^2 at (px,py), both normalized. Returns same value in all lanes.
__device__ float sample_cost(const float* feat, long cs, long ps,
                             const float* frefn, float px, float py, int lane) {
  float fx0 = fminf(fmaxf(floorf(px), 0.f), (float)(WW - 1));
  float fy0 = fminf(fmaxf(floorf(py), 0.f), (float)(HH - 1));
  int x0 = (int)fx0, y0 = (int)fy0;
  int x1 = min(x0 + 1, WW - 1), y1 = min(y0 + 1, HH - 1);
  float wx = px - fx0, wy = py - fy0;
  float w00 = (1.f-wx)*(1.f-wy), w01 = wx*(1.f-wy), w10 = (1.f-wx)*wy, w11 = wx*wy;
  float fr[4], rr[4];
  float n2 = 0.f, r2 = 0.f;
#pragma unroll
  for (int j = 0; j < 4; j++) {
    int c = lane + 32 * j;
    float v00 = fetchf(feat, cs, ps, c, y0, x0);
    float v01 = fetchf(feat, cs, ps, c, y0, x1);
    float v10 = fetchf(feat, cs, ps, c, y1, x0);
    float v11 = fetchf(feat, cs, ps, c, y1, x1);
    fr[j] = w00*v00 + w01*v01 + w10*v10 + w11*v11;
    rr[j] = frefn[c];
    n2 += fr[j]*fr[j];
    r2 += rr[j]*rr[j];
  }
  n2 = wave_sum(n2); r2 = wave_sum(r2);
  float invn = 1.f / fmaxf(sqrtf(n2), EPSF);
  float invr = 1.f / fmaxf(sqrtf(r2), EPSF);
  float cst = 0.f;
#pragma unroll
  for (int j = 0; j < 4; j++) {
    float d = fr[j]*invn - rr[j]*invr;
    cst += d*d;
  }
  return wave_sum(cst);
}

// -------- init --------
__global__ void init_kernel(const float* Rin, const float* tin, float* ws) {
  int t = threadIdx.x;
  if (t < 9) ws[WS_R + t] = Rin[t];
  if (t < 3) ws[WS_T + t] = tin[t];
  if (t == 0) { ws[WS_LAM] = 0.01f; ws[WS_COST] = 1e30f; }
  if (t < 6) ws[WS_JAC + t] = 0.f;
  for (int i = WS_H + t; i < 84; i += blockDim.x) ws[i] = 0.f;
}

// -------- CHW -> HWC transpose (feature map then fits in L2 with dense texels) --------
__global__ __launch_bounds__(256) void transpose_kernel(const float* __restrict__ in,
                                                        float* __restrict__ out) {
  int p = blockIdx.x * blockDim.x + threadIdx.x;
  if (p >= HH * WW) return;
  for (int c = 0; c < CC; c++)
    out[(long)p * CC + c] = in[(long)c * HH * WW + p];
}

// -------- main Gauss-Newton accumulation: H7 = A^T A via V_WMMA_F32_16X16X4_F32 --------
// LDS rows are 16 wide; cols 7..15 zeroed once per wave -> unconditional operand loads.
__global__ __launch_bounds__(256) void jtj_kernel(
    const float* __restrict__ p3D, const float* __restrict__ fref,
    const float* __restrict__ feat, long cs, long ps,
    const float* __restrict__ Km, float* __restrict__ ws, int N) {
  __shared__ float lds[8 * CC * 16];           // 8 waves x 128 rows x 16 cols = 64KB
  const int lane = threadIdx.x & 31;
  const int wv = threadIdx.x >> 5;
  float* A = &lds[wv * CC * 16];

  // zero the whole tile once; per point only cols 0..6 are rewritten
  for (int i = lane; i < CC * 16; i += 32) A[i] = 0.f;

  float Rm[9], tm[3];
#pragma unroll
  for (int i = 0; i < 9; i++) Rm[i] = ws[WS_R + i];
#pragma unroll
  for (int i = 0; i < 3; i++) tm[i] = ws[WS_T + i];
  float fx = Km[0], fy = Km[4], cx = Km[2], cy = Km[5];

  v8f acc = {0.f,0.f,0.f,0.f,0.f,0.f,0.f,0.f};
  float cnt = 0.f;

  int gw = blockIdx.x * (blockDim.x >> 5) + wv;
  int nw = gridDim.x * (blockDim.x >> 5);
  for (int n = gw; n < N; n += nw) {
    float X = p3D[3*n], Y = p3D[3*n+1], Z = p3D[3*n+2];
    float px, py, qx, qy, qz; int inim;
    proj_pt(Rm, tm, fx, fy, cx, cy, X, Y, Z, px, py, qx, qy, qz, inim);
    float vf = inim ? 1.f : 0.f;
    float invd = 1.f / qz;

    float fx0 = fminf(fmaxf(floorf(px), 0.f), (float)(WW - 1));
    float fy0 = fminf(fmaxf(floorf(py), 0.f), (float)(HH - 1));
    int x0 = (int)fx0, y0 = (int)fy0;
    int x1 = min(x0 + 1, WW - 1), y1 = min(y0 + 1, HH - 1);
    float wx = px - fx0, wy = py - fy0;
    float w00 = (1.f-wx)*(1.f-wy), w01 = wx*(1.f-wy), w10 = (1.f-wx)*wy, w11 = wx*wy;

    int prs[4] = { max(y0-1,0), y0, y1, min(y1+1, HH-1) };
    int pcs[4] = { max(x0-1,0), x0, x1, min(x1+1, WW-1) };

    float frw[4], gxs[4], gys[4], rr[4];
    float n2 = 0.f, r2 = 0.f;
#pragma unroll
    for (int j = 0; j < 4; j++) {
      int c = lane + 32 * j;
      float P[4][4];
#pragma unroll
      for (int a = 0; a < 4; a++)
#pragma unroll
        for (int b = 0; b < 4; b++)
          P[a][b] = fetchf(feat, cs, ps, c, prs[a], pcs[b]);
      float f00 = P[1][1], f01 = P[1][2], f10 = P[2][1], f11 = P[2][2];
      // Sobel (cross-correlation, edge-clamped) at the 4 bilinear corners
      float gx00 = ((P[0][2]-P[0][0]) + 2.f*(P[1][2]-P[1][0]) + (P[2][2]-P[2][0])) * 0.125f;
      float gx01 = ((P[0][3]-P[0][1]) + 2.f*(P[1][3]-P[1][1]) + (P[2][3]-P[2][1])) * 0.125f;
      float gx10 = ((P[1][2]-P[1][0]) + 2.f*(P[2][2]-P[2][0]) + (P[3][2]-P[3][0])) * 0.125f;
      float gx11 = ((P[1][3]-P[1][1]) + 2.f*(P[2][3]-P[2][1]) + (P[3][3]-P[3][1])) * 0.125f;
      float gy00 = ((P[2][0]-P[0][0]) + 2.f*(P[2][1]-P[0][1]) + (P[2][2]-P[0][2])) * 0.125f;
      float gy01 = ((P[2][1]-P[0][1]) + 2.f*(P[2][2]-P[0][2]) + (P[2][3]-P[0][3])) * 0.125f;
      float gy10 = ((P[3][0]-P[1][0]) + 2.f*(P[3][1]-P[1][1]) + (P[3][2]-P[1][2])) * 0.125f;
      float gy11 = ((P[3][1]-P[1][1]) + 2.f*(P[3][2]-P[1][2]) + (P[3][3]-P[1][3])) * 0.125f;
      frw[j] = w00*f00 + w01*f01 + w10*f10 + w11*f11;
      gxs[j] = w00*gx00 + w01*gx01 + w10*gx10 + w11*gx11;
      gys[j] = w00*gy00 + w01*gy01 + w10*gy10 + w11*gy11;
      rr[j] = fref[(long)n * CC + c];
      n2 += frw[j]*frw[j];
      r2 += rr[j]*rr[j];
    }
    n2 = wave_sum(n2); r2 = wave_sum(r2);
    float invn = 1.f / fmaxf(sqrtf(n2), EPSF);
    float invr = 1.f / fmaxf(sqrtf(r2), EPSF);
    float prx = 0.f, pry = 0.f;
#pragma unroll
    for (int j = 0; j < 4; j++) {
      float fj = frw[j] * invn;
      prx += fj * gxs[j];
      pry += fj * gys[j];
    }
    prx = wave_sum(prx); pry = wave_sum(pry);

    // J26 = Jpn @ [I | -skew(p3q)]
    float a = fx * invd, b = -fx * qx * invd * invd;
    float c2 = fy * invd, e = -fy * qy * invd * invd;
    float J0[6] = { a, 0.f, b, b*qy, a*qz - b*qx, -a*qy };
    float J1[6] = { 0.f, c2, e, e*qy - c2*qz, -e*qx, c2*qx };

#pragma unroll
    for (int j = 0; j < 4; j++) {
      int c = lane + 32 * j;
      float fj = frw[j] * invn;
      float Jfx = (gxs[j] - fj * prx) * invn;
      float Jfy = (gys[j] - fj * pry) * invn;
      int base = c * 16;
#pragma unroll
      for (int k = 0; k < 6; k++)
        A[base + k] = (Jfx * J0[k] + Jfy * J1[k]) * vf;
      A[base + 6] = (fj - rr[j] * invr) * vf;     // error column
    }
    asm volatile("s_wait_dscnt 0x0" ::: "memory");
    cnt += vf;

    // A^T A over 128 rows, K-chunks of 4. A-op (16x4, M=cols) and B-op (4x16, N=cols)
    // need the same (row, col=lane%16) element in the same lane/slot -> identical regs.
    int m = lane & 15;
    int khalf = (lane >> 4) << 1;   // 0 for lanes 0-15 (K=0,1), 2 for lanes 16-31 (K=2,3)
#pragma unroll 4
    for (int kc = 0; kc < 32; kc++) {
      int r0 = kc * 4 + khalf;
      float a0 = A[r0 * 16 + m];          // cols 7..15 are pre-zeroed
      float a1 = A[r0 * 16 + 16 + m];
      v2f av; av.x = a0; av.y = a1;
      acc = __builtin_amdgcn_wmma_f32_16x16x4_f32(false, av, false, av,
                                                  (short)0, acc, false, false);
    }
  }

  if (lane == 0) atomicAdd(&ws[WS_CNT], cnt);
  if (lane < 7) {                    // D[M=v][N=lane], lanes 0..6 / vgprs 0..6 hold 7x7
#pragma unroll
    for (int v = 0; v < 7; v++)
      atomicAdd(&ws[WS_H + v * 7 + lane], acc[v]);
  }
}

// -------- 1-thread: jacobi, scale, damp, 6x6 solve, so3exp, candidate pose --------
__global__ void solve_kernel(float* ws, int first) {
  if (threadIdx.x != 0) return;
  float H[7][7];
  for (int i = 0; i < 7; i++)
    for (int j = 0; j < 7; j++) H[i][j] = ws[WS_H + i * 7 + j];
  float den = ws[WS_CNT];
  if (first) {
    for (int k = 0; k < 6; k++)
      ws[WS_JAC + k] = 1.f / (1.f + sqrtf(fmaxf(H[k][k], 0.f)));
    ws[WS_COST] = H[6][6] / den;     // cost_best at iter 0
  }
  float jac[6];
  for (int k = 0; k < 6; k++) jac[k] = ws[WS_JAC + k];
  float M[6][7];
  for (int i = 0; i < 6; i++) {
    for (int j = 0; j < 6; j++) M[i][j] = H[i][j] * jac[i] * jac[j];
    M[i][6] = H[i][6] * jac[i];      // gradient
  }
  float lam = ws[WS_LAM];
  for (int i = 0; i < 6; i++) M[i][i] += (M[i][i] + 1e-9f) * lam;
  // Gauss-Jordan with partial pivoting
  for (int col = 0; col < 6; col++) {
    int piv = col; float best = fabsf(M[col][col]);
    for (int r = col + 1; r < 6; r++) {
      float v = fabsf(M[r][col]);
      if (v > best) { best = v; piv = r; }
    }
    if (piv != col)
      for (int c = 0; c < 7; c++) { float t = M[col][c]; M[col][c] = M[piv][c]; M[piv][c] = t; }
    float inv = 1.f / M[col][col];
    for (int r = 0; r < 6; r++) {
      if (r == col) continue;
      float f = M[r][col] * inv;
      for (int c = col; c < 7; c++) M[r][c] -= f * M[col][c];
    }
  }
  float delta[6];
  for (int i = 0; i < 6; i++) delta[i] = -(M[i][6] / M[i][i]) * jac[i];
  float wv0 = delta[3], wv1 = delta[4], wv2 = delta[5];
  float th2 = wv0*wv0 + wv1*wv1 + wv2*wv2;
  float th = sqrtf(th2);
  float Ac, Bc;
  if (th < 1e-7f) { Ac = 1.f; Bc = 0.5f; }
  else { Ac = sinf(th) / th; Bc = (1.f - cosf(th)) / th2; }
  float Wm[3][3] = {{0.f,-wv2,wv1},{wv2,0.f,-wv0},{-wv1,wv0,0.f}};
  float W2[3][3], dr[3][3];
  for (int i = 0; i < 3; i++)
    for (int j = 0; j < 3; j++) {
      float s = 0.f;
      for (int k = 0; k < 3; k++) s += Wm[i][k] * Wm[k][j];
      W2[i][j] = s;
    }
  for (int i = 0; i < 3; i++)
    for (int j = 0; j < 3; j++)
      dr[i][j] = ((i == j) ? 1.f : 0.f) + Ac * Wm[i][j] + Bc * W2[i][j];
  float Ro[9], to[3];
  for (int i = 0; i < 9; i++) Ro[i] = ws[WS_R + i];
  for (int i = 0; i < 3; i++) to[i] = ws[WS_T + i];
  for (int i = 0; i < 3; i++)
    for (int j = 0; j < 3; j++)
      ws[WS_RC + i * 3 + j] =
          dr[i][0] * Ro[0 + j] + dr[i][1] * Ro[3 + j] + dr[i][2] * Ro[6 + j];
  for (int i = 0; i < 3; i++)
    ws[WS_TC + i] = dr[i][0]*to[0] + dr[i][1]*to[1] + dr[i][2]*to[2] + delta[i];
  for (int i = 0; i < 49; i++) ws[WS_H + i] = 0.f;
  ws[WS_CNT] = 0.f; ws[WS_NUM] = 0.f; ws[WS_DEN] = 0.f;
}

// -------- trial-pose masked cost --------
__global__ __launch_bounds__(256) void eval_kernel(
    const float* __restrict__ p3D, const float* __restrict__ fref,
    const float* __restrict__ feat, long cs, long ps,
    const float* __restrict__ Km, float* __restrict__ ws, int N) {
  const int lane = threadIdx.x & 31;
  const int wv = threadIdx.x >> 5;
  float Rc[9], tc[3];
#pragma unroll
  for (int i = 0; i < 9; i++) Rc[i] = ws[WS_RC + i];
#pragma unroll
  for (int i = 0; i < 3; i++) tc[i] = ws[WS_TC + i];
  float fx = Km[0], fy = Km[4], cx = Km[2], cy = Km[5];
  float num = 0.f, den = 0.f;
  int gw = blockIdx.x * (blockDim.x >> 5) + wv;
  int nw = gridDim.x * (blockDim.x >> 5);
  for (int n = gw; n < N; n += nw) {
    float X = p3D[3*n], Y = p3D[3*n+1], Z = p3D[3*n+2];
    float px, py, qx, qy, qz; int inim;
    proj_pt(Rc, tc, fx, fy, cx, cy, X, Y, Z, px, py, qx, qy, qz, inim);
    float vf = inim ? 1.f : 0.f;
    float cst = sample_cost(feat, cs, ps, fref + (long)n * CC, px, py, lane);
    num += cst * vf;
    den += vf;
  }
  if (lane == 0) { atomicAdd(&ws[WS_NUM], num); atomicAdd(&ws[WS_DEN], den); }
}

// -------- accept/reject + lambda update --------
__global__ void accept_kernel(float* ws) {
  if (threadIdx.x != 0) return;
  float cost_new = ws[WS_NUM] / ws[WS_DEN];
  bool acc = (cost_new <= ws[WS_COST]);   // NaN -> reject, matches reference
  float lam = ws[WS_LAM] * (acc ? 0.1f : 10.f);
  ws[WS_LAM] = fminf(fmaxf(lam, 1e-8f), 1e4f);
  if (acc) {
    for (int i = 0; i < 9; i++) ws[WS_R + i] = ws[WS_RC + i];
    for (int i = 0; i < 3; i++) ws[WS_T + i] = ws[WS_TC + i];
    ws[WS_COST] = cost_new;
  }
  ws[WS_NUM] = 0.f; ws[WS_DEN] = 0.f;
}

// -------- outputs: R, t, cost_init (pose_init), cost_final (accepted pose) --------
__global__ __launch_bounds__(256) void final_kernel(
    const float* __restrict__ p3D, const float* __restrict__ fref,
    const float* __restrict__ feat, long cs, long ps,
    const float* __restrict__ Km, const float* __restrict__ Rin,
    const float* __restrict__ tin, const float* __restrict__ ws,
    float* __restrict__ out, int N) {
  const int lane = threadIdx.x & 31;
  const int wv = threadIdx.x >> 5;
  if (blockIdx.x == 0 && threadIdx.x < 12) out[threadIdx.x] = ws[threadIdx.x];
  float RA[9], tA[3], RB[9], tB[3];
#pragma unroll
  for (int i = 0; i < 9; i++) { RA[i] = Rin[i]; RB[i] = ws[WS_R + i]; }
#pragma unroll
  for (int i = 0; i < 3; i++) { tA[i] = tin[i]; tB[i] = ws[WS_T + i]; }
  float fx = Km[0], fy = Km[4], cx = Km[2], cy = Km[5];
  int gw = blockIdx.x * (blockDim.x >> 5) + wv;
  int nw = gridDim.x * (blockDim.x >> 5);
  for (int n = gw; n < N; n += nw) {
    float X = p3D[3*n], Y = p3D[3*n+1], Z = p3D[3*n+2];
    const float* frefn = fref + (long)n * CC;
    float px, py, qx, qy, qz; int i0, i1;
    proj_pt(RA, tA, fx, fy, cx, cy, X, Y, Z, px, py, qx, qy, qz, i0);
    float c0 = sample_cost(feat, cs, ps, frefn, px, py, lane);
    proj_pt(RB, tB, fx, fy, cx, cy, X, Y, Z, px, py, qx, qy, qz, i1);
    float c1 = sample_cost(feat, cs, ps, frefn, px, py, lane);
    if (lane == 0) {
      out[12 + n]     = i0 ? c0 : __builtin_nanf("");
      out[12 + N + n] = i1 ? c1 : __builtin_nanf("");
    }
  }
}

extern "C" void kernel_launch(void* const* d_in, const int* in_sizes, int n_in,
                              void* d_out, int out_size, void* d_ws, size_t ws_size,
                              hipStream_t stream) {
  const float* p3D  = (const float*)d_in[0];
  const float* fref = (const float*)d_in[1];
  const float* fq   = (const float*)d_in[2];
  const float* Km   = (const float*)d_in[3];
  const float* Rin  = (const float*)d_in[4];
  const float* tin  = (const float*)d_in[5];
  float* out = (float*)d_out;
  float* ws  = (float*)d_ws;
  int N = in_sizes[0] / 3;

  init_kernel<<<1, 64, 0, stream>>>(Rin, tin, ws);

  size_t needed = (size_t)(WS_FEAT + (size_t)HH * WW * CC) * sizeof(float);
  const float* feat;
  long cs, ps;
  if (ws_size >= needed) {
    transpose_kernel<<<(HH * WW + 255) / 256, 256, 0, stream>>>(fq, ws + WS_FEAT);
    feat = ws + WS_FEAT; cs = 1; ps = CC;            // HWC: dense channels per texel
  } else {
    feat = fq; cs = (long)HH * WW; ps = 1;           // CHW fallback
  }

  for (int it = 0; it < 8; ++it) {
    jtj_kernel<<<256, 256, 0, stream>>>(p3D, fref, feat, cs, ps, Km, ws, N);
    solve_kernel<<<1, 32, 0, stream>>>(ws, it == 0 ? 1 : 0);
    eval_kernel<<<256, 256, 0, stream>>>(p3D, fref, feat, cs, ps, Km, ws, N);
    accept_kernel<<<1, 32, 0, stream>>>(ws);
  }
  final_kernel<<<256, 256, 0, stream>>>(p3D, fref, feat, cs, ps, Km, Rin, tin,
                                        ws, out, N);
}